// BaGuaLLM_26079041421875
// MI455X (gfx1250) — compile-verified
//
#include <hip/hip_runtime.h>
#include <hip/hip_bf16.h>
#include <math.h>

typedef float v2f __attribute__((ext_vector_type(2)));
typedef float v8f __attribute__((ext_vector_type(8)));

#define NB 8
#define NS 2048
#define ND 1024
#define NH 8
#define NHD 128
#define NP 32

// ---------------------------------------------------------------------------
// GEMM1: heads[b,h,s,e] = (x @ W_proj^T)[m,n] * cos(pi*freqs[n]), m=b*S+s,
// n=h*HD+e.  Also accumulates hm[b,h,e] = mean_s heads (atomic, /2048 folded).
// One wave computes a 32(M) x 64(N) tile via V_WMMA_F32_16X16X4_F32:
// two 16-row M sub-tiles share the B fragments (0.75 loads per WMMA).
// ---------------------------------------------------------------------------
__global__ __launch_bounds__(256)
void bagua_gemm_heads(const float* __restrict__ x, const float* __restrict__ Wp,
                      const float* __restrict__ freqs,
                      float* __restrict__ heads, float* __restrict__ hm)
{
    const int wave  = ((int)blockIdx.x << 3) + ((int)threadIdx.x >> 5);
    const int lane  = (int)threadIdx.x & 31;
    const int half  = lane >> 4;        // K-pair selector
    const int l     = lane & 15;        // M (for A) / N (for B) within tile
    const int tileM = wave >> 4;        // 0..511  (32 rows each)
    const int n0    = (wave & 15) << 6; // 0..960 step 64

    const float* aRow0 = x  + (size_t)(tileM * 32 + l) * ND + (half << 1);
    const float* aRow1 = aRow0 + (size_t)16 * ND;
    const float* bRow  = Wp + (size_t)(n0 + l) * ND + (half << 1);

    v8f zero = {0.f,0.f,0.f,0.f,0.f,0.f,0.f,0.f};
    v8f acc[2][4] = {{zero, zero, zero, zero}, {zero, zero, zero, zero}};

    #pragma unroll 2
    for (int k = 0; k < ND; k += 4) {
        v2f a0 = *(const v2f*)(aRow0 + k);
        v2f a1 = *(const v2f*)(aRow1 + k);
        #pragma unroll
        for (int c = 0; c < 4; ++c) {
            v2f b = *(const v2f*)(bRow + (size_t)c * 16 * ND + k);
            acc[0][c] = __builtin_amdgcn_wmma_f32_16x16x4_f32(
                false, a0, false, b, (short)0, acc[0][c], false, false);
            acc[1][c] = __builtin_amdgcn_wmma_f32_16x16x4_f32(
                false, a1, false, b, (short)0, acc[1][c], false, false);
        }
    }

    #pragma unroll
    for (int u = 0; u < 2; ++u) {
        const int gs0 = tileM * 32 + u * 16;  // global row base (= b*S + s)
        const int bb  = gs0 >> 11;            // batch
        const int s0  = (gs0 & 2047) + (half << 3);
        #pragma unroll
        for (int c = 0; c < 4; ++c) {
            const int n = n0 + (c << 4) + l;
            const int h = n >> 7, e = n & 127;
            const float cs = __cosf(3.14159265358979323846f * freqs[n]);
            float psum = 0.f;
            float* hb = heads + ((size_t)(bb * NH + h) * NS + s0) * NHD + e;
            #pragma unroll
            for (int r = 0; r < 8; ++r) {
                float v = acc[u][c][r] * cs;
                hb[(size_t)r * NHD] = v;
                psum += v;
            }
            atomicAdd(&hm[(bb * NH + h) * NHD + e], psum * (1.0f / 2048.0f));
        }
    }
}

// ---------------------------------------------------------------------------
// Impedance: pv=tanh(hm@Wp^T+bp); pm=normalize(pv); dp=pm@pm^T;
// imp=softplus(gelu(dp*w1+b1)@w2+b2); coef=0.1/(1+imp), zero diagonal.
// One block per batch.
// ---------------------------------------------------------------------------
__global__ __launch_bounds__(256)
void bagua_impedance(const float* __restrict__ hm, const float* __restrict__ Wpol,
                     const float* __restrict__ bp, const float* __restrict__ w1,
                     const float* __restrict__ b1, const float* __restrict__ w2,
                     const float* __restrict__ b2, float* __restrict__ coef)
{
    const int b = blockIdx.x;
    const int t = threadIdx.x;
    __shared__ float s_hm[NH * NHD];
    __shared__ float s_pm[NH * NP];
    __shared__ float s_inv[NH];

    #pragma unroll
    for (int q = 0; q < 4; ++q)
        s_hm[t + 256 * q] = hm[b * NH * NHD + t + 256 * q];
    __syncthreads();

    const int h = t >> 5, p = t & 31;
    float a = bp[h * NP + p];
    for (int d = 0; d < NHD; ++d)
        a += s_hm[h * NHD + d] * Wpol[(h * NP + p) * NHD + d];
    const float pv = tanhf(a);
    s_pm[t] = pv;
    __syncthreads();

    if (t < NH) {
        float ss = 0.f;
        for (int q = 0; q < NP; ++q) { float u = s_pm[t * NP + q]; ss += u * u; }
        s_inv[t] = 1.0f / fmaxf(sqrtf(ss), 1e-12f);
    }
    __syncthreads();
    s_pm[t] = pv * s_inv[h];
    __syncthreads();

    if (t < NH * NH) {
        const int i = t >> 3, j = t & 7;
        float dp = 0.f;
        for (int q = 0; q < NP; ++q) dp += s_pm[i * NP + q] * s_pm[j * NP + q];
        float m = 0.f;
        for (int k = 0; k < 16; ++k) {
            float u = dp * w1[k] + b1[k];
            float g = 0.5f * u * (1.0f + erff(u * 0.70710678118654752f)); // exact gelu
            m += g * w2[k];
        }
        const float v = m + b2[0];
        const float sp = (v > 0.f) ? (v + log1pf(expf(-v))) : log1pf(expf(v));
        coef[b * 64 + t] = (i == j) ? 0.f : 0.1f / (1.0f + sp);
    }
}

// ---------------------------------------------------------------------------
// Mix: merged[b,s,i*HD+e] = heads[b,i,s,e] + (s+1)/S * sum_j coef[b,i,j]*heads[b,j,s,e]
// One block per (b,s); heads row staged through LDS once.
// ---------------------------------------------------------------------------
__global__ __launch_bounds__(256)
void bagua_mix(const float* __restrict__ heads, const float* __restrict__ coef,
               float* __restrict__ merged)
{
    const int row = blockIdx.x;           // b*S + s
    const int b = row >> 11, s = row & 2047;
    const int t = threadIdx.x;
    __shared__ float sh[NH * NHD];
    __shared__ float sc[NH * NH];

    #pragma unroll
    for (int q = 0; q < 4; ++q) {
        const int idx = t + 256 * q;
        const int j = idx >> 7, e = idx & 127;
        sh[idx] = heads[((size_t)(b * NH + j) * NS + s) * NHD + e];
    }
    if (t < 64) sc[t] = coef[b * 64 + t];
    __syncthreads();

    const float scale = (float)(s + 1) * (1.0f / 2048.0f);
    #pragma unroll
    for (int q = 0; q < 4; ++q) {
        const int n = t + 256 * q;
        const int i = n >> 7, e = n & 127;
        float tr = 0.f;
        #pragma unroll
        for (int j = 0; j < NH; ++j) tr += sc[i * NH + j] * sh[j * NHD + e];
        merged[(size_t)row * ND + n] = sh[n] + scale * tr;
    }
}

// ---------------------------------------------------------------------------
// GEMM2: out[m,n] = (merged @ Wo^T)[m,n] + bo[n] + x[m,n]   (residual fused)
// Same 32x64-per-wave WMMA structure as GEMM1.
// ---------------------------------------------------------------------------
__global__ __launch_bounds__(256)
void bagua_gemm_out(const float* __restrict__ merged, const float* __restrict__ Wo,
                    const float* __restrict__ bo, const float* __restrict__ x,
                    float* __restrict__ out)
{
    const int wave  = ((int)blockIdx.x << 3) + ((int)threadIdx.x >> 5);
    const int lane  = (int)threadIdx.x & 31;
    const int half  = lane >> 4;
    const int l     = lane & 15;
    const int tileM = wave >> 4;
    const int n0    = (wave & 15) << 6;

    const float* aRow0 = merged + (size_t)(tileM * 32 + l) * ND + (half << 1);
    const float* aRow1 = aRow0 + (size_t)16 * ND;
    const float* bRow  = Wo + (size_t)(n0 + l) * ND + (half << 1);

    v8f zero = {0.f,0.f,0.f,0.f,0.f,0.f,0.f,0.f};
    v8f acc[2][4] = {{zero, zero, zero, zero}, {zero, zero, zero, zero}};

    #pragma unroll 2
    for (int k = 0; k < ND; k += 4) {
        v2f a0 = *(const v2f*)(aRow0 + k);
        v2f a1 = *(const v2f*)(aRow1 + k);
        #pragma unroll
        for (int c = 0; c < 4; ++c) {
            v2f b = *(const v2f*)(bRow + (size_t)c * 16 * ND + k);
            acc[0][c] = __builtin_amdgcn_wmma_f32_16x16x4_f32(
                false, a0, false, b, (short)0, acc[0][c], false, false);
            acc[1][c] = __builtin_amdgcn_wmma_f32_16x16x4_f32(
                false, a1, false, b, (short)0, acc[1][c], false, false);
        }
    }

    #pragma unroll
    for (int u = 0; u < 2; ++u) {
        const int m0 = tileM * 32 + u * 16 + (half << 3);
        #pragma unroll
        for (int c = 0; c < 4; ++c) {
            const int n = n0 + (c << 4) + l;
            const float bias = bo[n];
            #pragma unroll
            for (int r = 0; r < 8; ++r) {
                const size_t idx = (size_t)(m0 + r) * ND + n;
                out[idx] = acc[u][c][r] + bias + x[idx];
            }
        }
    }
}

// ---------------------------------------------------------------------------
// LayerNorm over D=1024, in place on out. One block per row.
// ---------------------------------------------------------------------------
__global__ __launch_bounds__(256)
void bagua_layernorm(float* __restrict__ y, const float* __restrict__ g,
                     const float* __restrict__ bta)
{
    const int row = blockIdx.x;
    const int t = threadIdx.x;
    float* p = y + (size_t)row * ND;

    float v[4]; float s = 0.f, s2 = 0.f;
    #pragma unroll
    for (int q = 0; q < 4; ++q) {
        v[q] = p[t + 256 * q];
        s += v[q]; s2 += v[q] * v[q];
    }
    __shared__ float r1[256], r2[256];
    r1[t] = s; r2[t] = s2;
    __syncthreads();
    for (int off = 128; off > 0; off >>= 1) {
        if (t < off) { r1[t] += r1[t + off]; r2[t] += r2[t + off]; }
        __syncthreads();
    }
    const float mu  = r1[0] * (1.0f / ND);
    const float var = r2[0] * (1.0f / ND) - mu * mu;
    const float inv = rsqrtf(var + 1e-5f);
    #pragma unroll
    for (int q = 0; q < 4; ++q) {
        const int n = t + 256 * q;
        p[n] = (v[q] - mu) * inv * g[n] + bta[n];
    }
}

extern "C" void kernel_launch(void* const* d_in, const int* in_sizes, int n_in,
                              void* d_out, int out_size, void* d_ws, size_t ws_size,
                              hipStream_t stream)
{
    const float* x     = (const float*)d_in[0];
    const float* Wproj = (const float*)d_in[1];
    const float* freqs = (const float*)d_in[2];
    const float* Wpol  = (const float*)d_in[3];
    const float* bp    = (const float*)d_in[4];
    const float* w1    = (const float*)d_in[5];
    const float* b1    = (const float*)d_in[6];
    const float* w2    = (const float*)d_in[7];
    const float* b2    = (const float*)d_in[8];
    const float* Wo    = (const float*)d_in[9];
    const float* bo    = (const float*)d_in[10];
    const float* ln_g  = (const float*)d_in[11];
    const float* ln_b  = (const float*)d_in[12];
    float* out = (float*)d_out;

    float* heads  = (float*)d_ws;                      // B*H*S*HD
    float* hm     = heads  + (size_t)NB * NH * NS * NHD;
    float* coef   = hm     + (size_t)NB * NH * NHD;
    float* merged = coef   + (size_t)NB * NH * NH;

    hipMemsetAsync(hm, 0, sizeof(float) * NB * NH * NHD, stream);

    // 16384 rows / 32 per tile = 512 M-tiles; 16 N-groups of 64 -> 8192 waves
    bagua_gemm_heads<<<1024, 256, 0, stream>>>(x, Wproj, freqs, heads, hm);
    bagua_impedance <<<NB, 256, 0, stream>>>(hm, Wpol, bp, w1, b1, w2, b2, coef);
    bagua_mix       <<<NB * NS, 256, 0, stream>>>(heads, coef, merged);
    bagua_gemm_out  <<<1024, 256, 0, stream>>>(merged, Wo, bo, x, out);
    bagua_layernorm <<<NB * NS, 256, 0, stream>>>(out, ln_g, ln_b);

    (void)in_sizes; (void)n_in; (void)out_size; (void)ws_size;
}